// Bert_4604204942037
// MI455X (gfx1250) — compile-verified
//
#include <hip/hip_runtime.h>
#include <stdint.h>

// Ragged subword mean pooling, MI455X (gfx1250), wave32.
//   out[b,w,:] = mean(emb[b, st:ed, :]) if mask[b,w] && st < ed else 0
// B=32, S=1024, W=768, D=768.
//
// Roofline: reads 96 MB (spans are disjoint -> each embedding row read <= once),
// writes 72 MB, ~25 MFLOP => pure HBM stream, ~7.2 us @ 23.3 TB/s. No WMMA:
// a GEMM formulation (pooling-matrix x embedding) is a ~1000x FLOP inflation.
//
// CDNA5 data paths used:
//  * short spans (<=4 rows, common case): non-temporal global_load_b128 +
//    global_prefetch_b8, accumulate in VGPRs.
//  * long spans: Tensor Data Mover. st/ed are wave-uniform, so the span is a
//    wave-uniform 2D tile (768 x nr, stride 768) -> one tensor_load_to_lds
//    DMA per chunk, s_wait_tensorcnt, reduce from LDS via ds_load_b128.

typedef float v4f __attribute__((ext_vector_type(4)));
typedef unsigned int v4u __attribute__((ext_vector_type(4)));
typedef int v4i __attribute__((ext_vector_type(4)));
typedef int v8i __attribute__((ext_vector_type(8)));

#define B_ 32
#define S_ 1024
#define W_ 768
#define D_ 768
#define COLCH 6             // 6 chunks * 32 lanes * 4 floats = 768 columns
#define WAVES_PER_BLOCK 8   // 256 threads = 8 wave32
#define CHUNK_ROWS 4        // TDM staging chunk (rows of D floats)
                            // LDS: 8 waves * 4 * 3KB = 96 KB / block

// Issue TENSOR_LOAD_TO_LDS for a contiguous [nr x 768] f32 tile at gaddr,
// stride 768 elements, into LDS byte offset lds_off. D# per CDNA5 ISA ch.8.
// This toolchain exposes the 6-arg builtin:
//   (v4u group0, v8i group1, v4i group2, v4i group3, v8i extra, i32 cpol)
// Groups 2/3 (+extra) are only needed for >2D tensors; zero-filled here
// (tensor_dim2 = 0, tile_dim3 = 0 -> unused).
__device__ __forceinline__ void tdm_load_rows(uint32_t lds_off, uint64_t gaddr,
                                              int nr) {
    v4u g0;
    g0.x = 1u;                                    // [1:0] count=1 (valid user D#)
    g0.y = lds_off;                               // [63:32] lds_addr (bytes)
    g0.z = (uint32_t)gaddr;                       // [95:64] global_addr[31:0]
    g0.w = (uint32_t)((gaddr >> 32) & 0x01FFFFFFu)// [120:96] global_addr[56:32]
           | 0x80000000u;                         // [127:126] type=2 ("image")
    v8i g1;
    g1[0] = 0x00020000;                           // data_size=2 -> 4 bytes; mask=0
    g1[1] = (int)(768u << 16);                    // tensor_dim0[15:0]=768 (bits 63:48)
    g1[2] = (int)((uint32_t)nr << 16);            // tensor_dim1[15:0]=nr  (bits 95:80)
    g1[3] = (int)(768u << 16);                    // tile_dim0=768 (bits 127:112)
    g1[4] = nr;                                   // tile_dim1=nr, tile_dim2=0
    g1[5] = 768;                                  // tensor_dim0_stride[31:0]=768
    g1[6] = 0;                                    // stride hi / dim1_stride lo
    g1[7] = 0;                                    // dim1_stride hi (unused, 2D tile)
    __builtin_amdgcn_tensor_load_to_lds(g0, g1, (v4i)0, (v4i)0, (v8i)0, 0);
}

__global__ __launch_bounds__(256) void span_mean_kernel(
    const float* __restrict__ emb,   // [B, S, D]
    const int*   __restrict__ off,   // [B, W, 2]
    const int*   __restrict__ mask,  // [B, W]
    float*       __restrict__ out)   // [B, W, D]
{
    __shared__ float stage[WAVES_PER_BLOCK][CHUNK_ROWS * D_];

    const int wave = (int)(threadIdx.x >> 5);
    const int lane = (int)(threadIdx.x & 31u);
    const int g    = (int)blockIdx.x * WAVES_PER_BLOCK + wave;  // flat (b*W + w)
    if (g >= B_ * W_) return;

    const int b  = g / W_;
    const int st = off[2 * g + 0];
    const int ed = off[2 * g + 1];
    const int m  = mask[g];

    v4f acc[COLCH];
#pragma unroll
    for (int c = 0; c < COLCH; ++c) acc[c] = (v4f)0.0f;

    const int  len   = ed - st;
    const bool valid = (m > 0) && (len > 0);
    if (valid) {
        const float* gbase = emb + (size_t)b * (S_ * (size_t)D_)
                                 + (size_t)st * D_;
        if (len <= CHUNK_ROWS) {
            // Short span (common): direct NT b128 loads, VGPR accumulation.
            for (int r = 0; r < len; ++r) {
                const float* row = gbase + (size_t)r * D_;
                if (r + 1 < len) __builtin_prefetch(row + D_, 0, 0);
#pragma unroll
                for (int c = 0; c < COLCH; ++c) {
                    v4f v = __builtin_nontemporal_load(
                        (const v4f*)(row + c * 128 + lane * 4));
                    acc[c] += v;
                }
            }
        } else {
            // Long span: TDM bulk DMA into this wave's LDS staging buffer,
            // CHUNK_ROWS rows per descriptor, then reduce from LDS.
            // Flat->LDS: low 32 bits of a __shared__ pointer are the
            // workgroup-relative LDS byte offset.
            const uint32_t lds_off = (uint32_t)(uintptr_t)&stage[wave][0];
            int r = 0;
            while (r < len) {
                const int nr = (len - r < CHUNK_ROWS) ? (len - r) : CHUNK_ROWS;
                tdm_load_rows(lds_off,
                              (uint64_t)(uintptr_t)(gbase + (size_t)r * D_), nr);
                __builtin_amdgcn_s_wait_tensorcnt(0);  // DMA -> LDS visible
                for (int rr = 0; rr < nr; ++rr) {
#pragma unroll
                    for (int c = 0; c < COLCH; ++c) {
                        // ds_load_b128 from the staged tile. The accumulate
                        // consumes the data (DScnt waited) before the next
                        // tensor_load can overwrite this buffer.
                        v4f v = *(const v4f*)&stage[wave][rr * D_ + c * 128 +
                                                         lane * 4];
                        acc[c] += v;
                    }
                }
                r += nr;
            }
        }
        const float inv = 1.0f / (float)len;
#pragma unroll
        for (int c = 0; c < COLCH; ++c) acc[c] *= inv;
    }
    // Invalid spans fall through with acc == 0 -> writes the required zeros.

    float* orow = out + (size_t)g * D_;
#pragma unroll
    for (int c = 0; c < COLCH; ++c) {
        __builtin_nontemporal_store(acc[c], (v4f*)(orow + c * 128 + lane * 4));
    }
}

extern "C" void kernel_launch(void* const* d_in, const int* in_sizes, int n_in,
                              void* d_out, int out_size, void* d_ws, size_t ws_size,
                              hipStream_t stream) {
    const float* emb  = (const float*)d_in[0];  // [B,S,D] float32
    const int*   off  = (const int*)d_in[1];    // [B,W,2] int32
    const int*   mask = (const int*)d_in[2];    // [B,W]   int32
    float*       out  = (float*)d_out;          // [B,W,D] float32

    const int rows = B_ * W_;                       // 24576 output rows
    dim3 grid(rows / WAVES_PER_BLOCK);              // 3072 blocks
    dim3 block(256);                                // 8 waves of 32
    span_mean_kernel<<<grid, block, 0, stream>>>(emb, off, mask, out);
}